// Valueblock_37623913513624
// MI455X (gfx1250) — compile-verified
//
#include <hip/hip_runtime.h>
#include <hip/hip_bf16.h>

typedef float v2f __attribute__((ext_vector_type(2)));
typedef float v8f __attribute__((ext_vector_type(8)));

#define T_LEN   1024
#define D_HALF  256
#define NLABEL  64
#define KDIM    (NLABEL * D_HALF)   // 16384
#define CDIM    256
#define MPAD    16                  // WMMA M dimension (rows 0..3 live, 4..15 zero)

// ---------------------------------------------------------------------------
// Phase 1: acc[b, l*256+d] = sum_{t: label[b,t]==l} scores[b,t]*weight[idx,half_off+d]
// Grid = 16 blocks: blocks 0..3 accumulate (one per batch b), blocks 4..15
// zero-fill the pad rows so phase 2 needs no predication on A loads.
// Thread d exclusively owns LDS addresses {l*256+d}: conflict-free,
// atomic-free, deterministic.
// ---------------------------------------------------------------------------
__global__ void __launch_bounds__(256)
value_acc_kernel(const int* __restrict__ indices,
                 const float* __restrict__ scores,
                 const int* __restrict__ label,
                 const int* __restrict__ index_p,
                 const float* __restrict__ weight,
                 float* __restrict__ acc_ws)
{
    const int b = blockIdx.x;

    if (b >= 4) {                       // pad rows 4..15 -> zero
        float* dst = acc_ws + (size_t)b * KDIM;
        #pragma unroll
        for (int i = 0; i < KDIM / 256; ++i)
            dst[i * 256 + threadIdx.x] = 0.0f;
        return;
    }

    __shared__ float lacc[NLABEL * D_HALF];   // 64 KB
    const int d = threadIdx.x;
    const int half_off = (*index_p == 1) ? D_HALF : 0;

    #pragma unroll
    for (int l = 0; l < NLABEL; ++l)
        lacc[l * D_HALF + d] = 0.0f;
    __syncthreads();

    const int base = b * T_LEN;
    for (int t = 0; t < T_LEN; ++t) {
        const int   idx = indices[base + t];
        const int   lbl = label[base + t];
        const float s   = scores[base + t];
        // prefetch a future gathered row into cache (global_prefetch_b8)
        if (t + 8 < T_LEN) {
            const int pidx = indices[base + t + 8];
            __builtin_prefetch(&weight[(size_t)pidx * 512 + half_off + d], 0, 1);
        }
        const float val = weight[(size_t)idx * 512 + half_off + d];
        lacc[lbl * D_HALF + d] += s * val;
    }
    __syncthreads();

    float* dst = acc_ws + (size_t)b * KDIM;
    #pragma unroll
    for (int l = 0; l < NLABEL; ++l)
        dst[l * D_HALF + d] = lacc[l * D_HALF + d];
}

// ---------------------------------------------------------------------------
// Phase 2: out[b,c] = sum_k acc[b,k] * Wflat[k,c],  k = l*256+d,
//          Wflat[k,c] = W[l, half_off + d, c]   (W is (64, 512, 256) row-major)
// M=16 (rows 0..3 live), K=16384, N=256.  16 blocks (N tiles of 16) x 8 waves.
// Each wave handles 8 labels (K-slice 2048) -> 512 V_WMMA_F32_16X16X4_F32,
// with affine per-label addressing so the unrolled loop uses immediate
// offsets, then the 8 per-wave accumulators reduce through LDS.
// ---------------------------------------------------------------------------
__global__ void __launch_bounds__(256)
value_gemm_kernel(const float* __restrict__ acc,
                  const float* __restrict__ W,
                  const int* __restrict__ index_p,
                  float* __restrict__ out)
{
    const int half_off = (*index_p == 1) ? D_HALF : 0;
    const int n0   = blockIdx.x * 16;
    const int lane = threadIdx.x & 31;
    const int wave = threadIdx.x >> 5;
    const int m    = lane & 15;            // A row / B column within tile
    const int kb   = (lane >> 4) << 1;     // this lane covers K pair {kb, kb+1}

    v8f c = {};

    // A fragment base: row m, this wave's K-slice, lane's K pair.
    const float* arow = acc + (size_t)m * KDIM + wave * (8 * D_HALF) + kb;
    // B fragment base: column n0+m of W_h, shifted by lane's K pair (kb rows).
    const float* bcol = W + ((size_t)half_off + kb) * CDIM + n0 + m;

    #pragma unroll 1
    for (int j = 0; j < 8; ++j) {
        const int    l  = wave * 8 + j;
        const float* ap = arow + j * D_HALF;
        const float* bp = bcol + (size_t)l * 512 * CDIM;
        #pragma unroll 8
        for (int d = 0; d < D_HALF; d += 4) {
            v2f a = *(const v2f*)(ap + d);           // global_load_b64
            v2f bf;
            bf.x = bp[(size_t)d * CDIM];             // imm-offset b32 loads
            bf.y = bp[(size_t)(d + 1) * CDIM];
            // D = A(16x4,f32) * B(4x16,f32) + C  -> v_wmma_f32_16x16x4_f32
            c = __builtin_amdgcn_wmma_f32_16x16x4_f32(
                    /*neg_a=*/false, a, /*neg_b=*/false, bf,
                    /*c_mod=*/(short)0, c, /*reuse_a=*/false, /*reuse_b=*/false);
        }
    }

    // C fragment: VGPR r, lanes 0-15 hold M=r, N=lane. Rows 0..3 are live.
    __shared__ float red[8][4][16];
    if (lane < 16) {
        red[wave][0][lane] = c[0];
        red[wave][1][lane] = c[1];
        red[wave][2][lane] = c[2];
        red[wave][3][lane] = c[3];
    }
    __syncthreads();

    if (threadIdx.x < 64) {
        const int bb = threadIdx.x >> 4;
        const int nn = threadIdx.x & 15;
        float s = 0.0f;
        #pragma unroll
        for (int w = 0; w < 8; ++w) s += red[w][bb][nn];
        out[bb * CDIM + n0 + nn] = s;
    }
}

// ---------------------------------------------------------------------------
// Inputs (setup_inputs order): 0=indices(i32), 1=scores(f32), 2=W(f32),
// 3=label(i32), 4=index(i32 scalar), 5=weight(f32).  Output: (4,256) f32.
// Workspace: 16 * 16384 floats = 1 MB padded accumulator.
// ---------------------------------------------------------------------------
extern "C" void kernel_launch(void* const* d_in, const int* in_sizes, int n_in,
                              void* d_out, int out_size, void* d_ws, size_t ws_size,
                              hipStream_t stream) {
    const int*   indices = (const int*)  d_in[0];
    const float* scores  = (const float*)d_in[1];
    const float* W       = (const float*)d_in[2];
    const int*   label   = (const int*)  d_in[3];
    const int*   index_p = (const int*)  d_in[4];
    const float* weight  = (const float*)d_in[5];
    float*       out     = (float*)d_out;
    float*       acc_ws  = (float*)d_ws;          // MPAD * KDIM floats = 1 MB

    value_acc_kernel<<<MPAD, 256, 0, stream>>>(indices, scores, label, index_p,
                                               weight, acc_ws);
    value_gemm_kernel<<<16, 256, 0, stream>>>(acc_ws, W, index_p, out);
}